// Fourier3dFunctionPy_57629871178215
// MI455X (gfx1250) — compile-verified
//
#include <hip/hip_runtime.h>
#include <hip/hip_bf16.h>

typedef __attribute__((ext_vector_type(2))) float v2f;
typedef __attribute__((ext_vector_type(8))) float v8f;

#define NV    512
#define NF    1024
#define TILE  16
#define NPTS  32768
#define SEPS  3.4526698e-4f   // sqrt(1.19209e-07)

// ---------------- Prep: V = Verts - box[:,0]; trisar = 2*area/meshar ----------------
__launch_bounds__(1024)
__global__ void fourier_prep(const float* __restrict__ Verts,
                             const float* __restrict__ box,
                             const int*  __restrict__ faces,
                             float* __restrict__ Vws,    // 512*3
                             float* __restrict__ tris)   // 1024
{
    __shared__ float sV[NV * 3];
    __shared__ float sarr[NF];
    const int t = threadIdx.x;

    for (int i = t; i < NV * 3; i += 1024) {
        float v = Verts[i] - box[(i % 3) * 2];
        sV[i] = v;
        Vws[i] = v;
    }
    __syncthreads();

    const int fa = faces[3 * t + 0], fb = faces[3 * t + 1], fc = faces[3 * t + 2];
    const float ax = sV[3 * fa], ay = sV[3 * fa + 1], az = sV[3 * fa + 2];
    const float bx = sV[3 * fb], by = sV[3 * fb + 1], bz = sV[3 * fb + 2];
    const float cx = sV[3 * fc], cy = sV[3 * fc + 1], cz = sV[3 * fc + 2];
    const float e1x = bx - ax, e1y = by - ay, e1z = bz - az;
    const float e2x = cx - ax, e2y = cy - ay, e2z = cz - az;
    const float crx = e1y * e2z - e1z * e2y;
    const float cry = e1z * e2x - e1x * e2z;
    const float crz = e1x * e2y - e1y * e2x;
    const float ar = 0.5f * sqrtf(crx * crx + cry * cry + crz * crz);

    sarr[t] = ar;
    __syncthreads();
    for (int s = NF / 2; s > 0; s >>= 1) {
        if (t < s) sarr[t] += sarr[t + s];
        __syncthreads();
    }
    const float meshar = sarr[0];
    tris[t] = ar * 2.0f / meshar;
}

// ---------------- Main: 16 Xi rows per block ----------------
__launch_bounds__(256)
__global__ void fourier_main(const float* __restrict__ xi0,
                             const float* __restrict__ xi1,
                             const float* __restrict__ xi2,
                             const int*  __restrict__ faces,
                             const float* __restrict__ Vws,
                             const float* __restrict__ tris,
                             float* __restrict__ out)    // interleaved complex, 2*NPTS
{
    __shared__ float  xiv_s[TILE * NV];      // 32 KB
    __shared__ float2 cs_s[TILE * NV];       // 64 KB (cos, sin)

    const int tid  = threadIdx.x;
    const int wave = tid >> 5;
    const int lane = tid & 31;
    const int half = lane >> 4;
    const int lg   = lane & 15;
    const int base = blockIdx.x * TILE;

    // ---- Phase 1: xiv[16][512] tile via V_WMMA_F32_16X16X4_F32 ----
    // A (16x4): row M = lg; half0 -> (K0,K1) = (xi_x, xi_y); half1 -> (K2,K3) = (xi_z, 0)
    {
        const int gm = base + lg;
        const float xix = xi0[(gm >> 10) & 31];
        const float xiy = xi1[(gm >> 5) & 31];
        const float xiz = xi2[gm & 31];
        v2f a;
        a.x = half ? xiz : xix;
        a.y = half ? 0.0f : xiy;

        for (int c = wave; c < NV / 16; c += 8) {
            const int n = c * 16 + lg;
            const float vx = Vws[3 * n + 0];
            const float vy = Vws[3 * n + 1];
            const float vz = Vws[3 * n + 2];
            // B (4x16): col N = lg; half0 -> rows (K0,K1) = (Vx,Vy); half1 -> (Vz,0)
            v2f b;
            b.x = half ? vz : vx;
            b.y = half ? 0.0f : vy;

            v8f acc = {0.f, 0.f, 0.f, 0.f, 0.f, 0.f, 0.f, 0.f};
            acc = __builtin_amdgcn_wmma_f32_16x16x4_f32(
                /*neg_a=*/false, a, /*neg_b=*/false, b,
                /*c_mod=*/(short)0, acc, /*reuse_a=*/false, /*reuse_b=*/false);

            // C/D layout: VGPR r, lanes0-15 -> M=r, lanes16-31 -> M=r+8; N = lg
#pragma unroll
            for (int r = 0; r < 8; ++r)
                xiv_s[(r + 8 * half) * NV + c * 16 + lg] = acc[r];
        }
    }
    __syncthreads();

    // ---- Phase 2: per-(row, vertex) cos/sin table ----
    for (int idx = tid; idx < TILE * NV; idx += 256) {
        float s, c;
        __sincosf(xiv_s[idx], &s, &c);
        cs_s[idx] = make_float2(c, s);
    }
    __syncthreads();

    // ---- Phase 3: faces loop; each 16-lane group owns one Xi row ----
    const int row = wave * 2 + half;
    const float*  xr = &xiv_s[row * NV];
    const float2* cr = &cs_s[row * NV];

    float accRe = 0.f, accIm = 0.f;
    for (int f = lg; f < NF; f += 16) {
        const int fa = faces[3 * f + 0];
        const int fb = faces[3 * f + 1];
        const int fc = faces[3 * f + 2];
        const float xia = xr[fa], xib = xr[fb], xic = xr[fc];
        const float2 ea = cr[fa], eb = cr[fb], ec = cr[fc];

        const float xiab = xib - xia, xibc = xic - xib, xica = xia - xic;
        const float aab = fabsf(xiab), abc = fabsf(xibc), aca = fabsf(xica);
        const bool s2 = aab < SEPS, s3 = abc < SEPS, s4 = aca < SEPS;
        const bool c2 = aab <= SEPS, c3 = abc <= SEPS, c4 = aca <= SEPS;
        const bool R1 = (s2 && s3) || (s3 && s4) || (s4 && s2);
        const bool R2 = c2 && !R1;
        const bool R3 = c3 && !R2 && !R1;
        const bool R4 = c4 && !R3 && !R2 && !R1;

        float rre, rim;
        if (R1) {
            // exp(-i*(xia+xib+xic)/3)/2
            const float m = (xia + xib + xic) * (1.0f / 3.0f);
            float sm, cm;
            __sincosf(m, &sm, &cm);
            rre = 0.5f * cm;
            rim = -0.5f * sm;
        } else if (R2 || R3 || R4) {
            // pair case: (i*em + (em - e3)/d)/d, em = exp(-i*m), m = (x1+x2)/2, d = m - x3
            float x1, x2, x3;
            float2 e3;
            if (R2)      { x1 = xia; x2 = xib; x3 = xic; e3 = ec; }
            else if (R3) { x1 = xib; x2 = xic; x3 = xia; e3 = ea; }
            else         { x1 = xic; x2 = xia; x3 = xib; e3 = eb; }
            const float m = 0.5f * (x1 + x2);
            const float d = m - x3;
            float sm, cm;
            __sincosf(m, &sm, &cm);
            const float invd = 1.0f / d;
            rre = (sm + (cm - e3.x) * invd) * invd;
            rim = (cm + (e3.y - sm) * invd) * invd;
        } else {
            // R5: exp(-i xia)/(xiab*xica) + exp(-i xib)/(xibc*xiab) + exp(-i xic)/(xica*xibc)
            const float ida = 1.0f / (xiab * xica);
            const float idb = 1.0f / (xibc * xiab);
            const float idc = 1.0f / (xica * xibc);
            rre =  (ea.x * ida + eb.x * idb + ec.x * idc);
            rim = -(ea.y * ida + eb.y * idb + ec.y * idc);
        }
        const float t = tris[f];
        accRe += t * rre;
        accIm += t * rim;
    }

    // reduce across the 16-lane group (wave32 xor shuffles stay inside the group)
    for (int m = 8; m >= 1; m >>= 1) {
        accRe += __shfl_xor(accRe, m, 32);
        accIm += __shfl_xor(accIm, m, 32);
    }
    if (lg == 0) {
        const int g = base + row;
        out[2 * g + 0] = accRe;
        out[2 * g + 1] = accIm;
    }
}

extern "C" void kernel_launch(void* const* d_in, const int* in_sizes, int n_in,
                              void* d_out, int out_size, void* d_ws, size_t ws_size,
                              hipStream_t stream) {
    const float* Verts = (const float*)d_in[0];   // 512*3
    const float* box   = (const float*)d_in[1];   // 3*2
    const float* xi0   = (const float*)d_in[2];   // 32
    const float* xi1   = (const float*)d_in[3];   // 32
    const float* xi2   = (const float*)d_in[4];   // 32
    const int*   faces = (const int*)d_in[5];     // 1024*3

    float* Vws  = (float*)d_ws;       // 1536 floats
    float* tris = Vws + NV * 3;       // 1024 floats
    float* out  = (float*)d_out;      // 2*32768 floats (complex interleaved)

    fourier_prep<<<1, 1024, 0, stream>>>(Verts, box, faces, Vws, tris);
    fourier_main<<<NPTS / TILE, 256, 0, stream>>>(xi0, xi1, xi2, faces, Vws, tris, out);
}